// GCN_21509196218553
// MI455X (gfx1250) — compile-verified
//
#include <hip/hip_runtime.h>
#include <hip/hip_bf16.h>

typedef float v2f __attribute__((ext_vector_type(2)));
typedef float v8f __attribute__((ext_vector_type(8)));

#define HID 128
#define DENSE_DIM 64

// ---------------- degree / norm precompute ----------------

__global__ void k_deg(const int* __restrict__ dst, int* __restrict__ degi, int E) {
    int e = blockIdx.x * 256 + threadIdx.x;
    if (e < E) atomicAdd(&degi[dst[e]], 1);
}

__global__ void k_dis(const int* __restrict__ degi, float* __restrict__ dis, int N) {
    int n = blockIdx.x * 256 + threadIdx.x;
    if (n < N) dis[n] = rsqrtf((float)(degi[n] + 1));   // +1 for self-loop
}

__global__ void k_norm(const int* __restrict__ ei, const float* __restrict__ dis,
                       float* __restrict__ norm, int E) {
    int e = blockIdx.x * 256 + threadIdx.x;
    if (e < E) norm[e] = dis[ei[e]] * dis[ei[E + e]];
}

// ---------------- layer 0: x(N,3) @ W0(3,128), float4 per thread ----------------

__global__ void k_lin3(const float* __restrict__ x, const float* __restrict__ W,
                       float* __restrict__ T, int N) {
    size_t i = (size_t)blockIdx.x * 256 + threadIdx.x;      // one float4 of output
    if (i >= (size_t)N * (HID / 4)) return;
    size_t n = i >> 5;
    int f4 = (int)(i & 31) * 4;
    const float* xr = x + n * 3;
    float x0 = xr[0], x1 = xr[1], x2 = xr[2];
    float4 w0 = *(const float4*)(W + f4);
    float4 w1 = *(const float4*)(W + HID + f4);
    float4 w2 = *(const float4*)(W + 2 * HID + f4);
    float4 o;
    o.x = x0 * w0.x + x1 * w1.x + x2 * w2.x;
    o.y = x0 * w0.y + x1 * w1.y + x2 * w2.y;
    o.z = x0 * w0.z + x1 * w1.z + x2 * w2.z;
    o.w = x0 * w0.w + x1 * w1.w + x2 * w2.w;
    *(float4*)(T + i * 4) = o;
}

// ---------------- fp32 WMMA GEMM: T = H(N,128) @ W(128,128) ----------------
// 256 threads = 8 waves; wave w computes rows [blk*128 + 16w, +16) x all 128 cols.

__global__ __launch_bounds__(256) void k_gemm128_wmma(const float* __restrict__ H,
                                                      const float* __restrict__ W,
                                                      float* __restrict__ T, int N) {
    __shared__ float sW[HID * HID];                    // 64 KB
    const int tid = threadIdx.x;
    {   // cooperative stage of W into LDS
        const float4* Wv = (const float4*)W;
        float4* sWv = (float4*)sW;
        #pragma unroll
        for (int i = 0; i < (HID * HID / 4) / 256; ++i)
            sWv[tid + i * 256] = Wv[tid + i * 256];
    }
    __syncthreads();

    const int wave = tid >> 5;
    const int lane = tid & 31;
    const int m  = lane & 15;      // row within tile (A) / col within tile (B,D)
    const int kh = lane >> 4;      // K-half selector
    const int rowBase = blockIdx.x * 128 + wave * 16;

    int rowA = rowBase + m;
    if (rowA >= N) rowA = N - 1;                       // clamp; stores are guarded
    const float* hrow = H + (size_t)rowA * HID;

    v8f acc[8];
    #pragma unroll
    for (int t = 0; t < 8; ++t)
        #pragma unroll
        for (int r = 0; r < 8; ++r) acc[t][r] = 0.0f;

    #pragma unroll 2
    for (int k0 = 0; k0 < HID; k0 += 4) {
        v2f a;                                          // A[m][k0+2kh .. +1], contiguous b64
        a.x = hrow[k0 + kh * 2 + 0];
        a.y = hrow[k0 + kh * 2 + 1];
        #pragma unroll
        for (int t = 0; t < 8; ++t) {
            v2f b;                                      // B[k0+2kh][t*16+m], B[k0+2kh+1][...]
            b.x = sW[(k0 + kh * 2 + 0) * HID + t * 16 + m];
            b.y = sW[(k0 + kh * 2 + 1) * HID + t * 16 + m];
            acc[t] = __builtin_amdgcn_wmma_f32_16x16x4_f32(
                false, a, false, b, (short)0, acc[t], false, false);
        }
    }

    // D layout: VGPR r -> row (rowBase + kh*8 + r), col = t*16 + m
    if (rowBase + 16 <= N) {
        // fast path: whole 16-row strip in range (all blocks except the last)
        float* Trow = T + (size_t)(rowBase + kh * 8) * HID + m;
        #pragma unroll
        for (int t = 0; t < 8; ++t)
            #pragma unroll
            for (int r = 0; r < 8; ++r)
                Trow[(size_t)r * HID + t * 16] = acc[t][r];
    } else {
        #pragma unroll
        for (int t = 0; t < 8; ++t)
            #pragma unroll
            for (int r = 0; r < 8; ++r) {
                int orow = rowBase + kh * 8 + r;
                if (orow < N) T[(size_t)orow * HID + t * 16 + m] = acc[t][r];
            }
    }
}

// ---------------- edge scatter: agg[dst] += T[src] * norm ----------------
// one wave per edge, 4 floats per lane (128 features)

__global__ __launch_bounds__(256) void k_scatter(const int* __restrict__ src,
                                                 const int* __restrict__ dst,
                                                 const float* __restrict__ norm,
                                                 const float* __restrict__ T,
                                                 float* __restrict__ agg, int E) {
    int e = blockIdx.x * 8 + (threadIdx.x >> 5);
    if (e >= E) return;
    int lane = threadIdx.x & 31;
    int s = src[e], d = dst[e];
    float nrm = norm[e];
    const float4 v = *(const float4*)(T + (size_t)s * HID + lane * 4);
    float* o = agg + (size_t)d * HID + lane * 4;
    unsafeAtomicAdd(o + 0, v.x * nrm);
    unsafeAtomicAdd(o + 1, v.y * nrm);
    unsafeAtomicAdd(o + 2, v.z * nrm);
    unsafeAtomicAdd(o + 3, v.w * nrm);
}

// ---------------- finalize: agg += self-loop + bias, relu (in place, float4) ----------------

__global__ void k_finalize(float* __restrict__ agg, const float* __restrict__ T,
                           const float* __restrict__ dis, const float* __restrict__ b,
                           int N) {
    size_t i = (size_t)blockIdx.x * 256 + threadIdx.x;      // one float4
    if (i >= (size_t)N * (HID / 4)) return;
    size_t n = i >> 5;
    int f4 = (int)(i & 31) * 4;
    float dn = dis[n];
    float sl = dn * dn;
    size_t base = i * 4;
    float4 a  = *(const float4*)(agg + base);
    float4 t  = *(const float4*)(T + base);
    float4 bb = *(const float4*)(b + f4);
    float4 o;
    o.x = a.x + t.x * sl + bb.x;
    o.y = a.y + t.y * sl + bb.y;
    o.z = a.z + t.z * sl + bb.z;
    o.w = a.w + t.w * sl + bb.w;
    o.x = o.x > 0.0f ? o.x : 0.0f;
    o.y = o.y > 0.0f ? o.y : 0.0f;
    o.z = o.z > 0.0f ? o.z : 0.0f;
    o.w = o.w > 0.0f ? o.w : 0.0f;
    *(float4*)(agg + base) = o;
}

// ---------------- pooling + MLP head ----------------

__global__ void k_pool_cnt(const int* __restrict__ batch, int* __restrict__ cnt, int N) {
    int n = blockIdx.x * 256 + threadIdx.x;
    if (n < N) atomicAdd(&cnt[batch[n]], 1);
}

__global__ void k_pool_sum(const int* __restrict__ batch, const float* __restrict__ H,
                           float* __restrict__ pooled, int N) {
    size_t i = (size_t)blockIdx.x * 256 + threadIdx.x;      // one float4
    if (i >= (size_t)N * (HID / 4)) return;
    size_t n = i >> 5;
    int f4 = (int)(i & 31) * 4;
    float4 h = *(const float4*)(H + i * 4);
    float* o = pooled + (size_t)batch[n] * HID + f4;
    unsafeAtomicAdd(o + 0, h.x);
    unsafeAtomicAdd(o + 1, h.y);
    unsafeAtomicAdd(o + 2, h.z);
    unsafeAtomicAdd(o + 3, h.w);
}

__global__ void k_dense1(const float* __restrict__ pooled, const int* __restrict__ cnt,
                         const float* __restrict__ W, const float* __restrict__ b,
                         float* __restrict__ o, int G) {
    int i = blockIdx.x * 256 + threadIdx.x;
    if (i >= G * DENSE_DIM) return;
    int g = i >> 6, j = i & (DENSE_DIM - 1);
    float inv = 1.0f / fmaxf((float)cnt[g], 1.0f);
    const float* p = pooled + (size_t)g * HID;
    float s = b[j];
    #pragma unroll 4
    for (int f = 0; f < HID; ++f) s += (p[f] * inv) * W[f * DENSE_DIM + j];
    o[i] = s > 0.0f ? s : 0.0f;
}

__global__ void k_dense2(const float* __restrict__ d1, const float* __restrict__ W,
                         const float* __restrict__ b, float* __restrict__ out, int G) {
    int g = blockIdx.x * 256 + threadIdx.x;
    if (g >= G) return;
    const float* r = d1 + (size_t)g * DENSE_DIM;
    float s = b[0];
    #pragma unroll 4
    for (int j = 0; j < DENSE_DIM; ++j) s += r[j] * W[j];
    out[g] = s;
}

// ---------------- launch ----------------

extern "C" void kernel_launch(void* const* d_in, const int* in_sizes, int n_in,
                              void* d_out, int out_size, void* d_ws, size_t ws_size,
                              hipStream_t stream) {
    const float* x   = (const float*)d_in[0];
    const int*   ei  = (const int*)d_in[1];   // [2,E]: src=ei[0..E), dst=ei[E..2E)
    const int*   bat = (const int*)d_in[2];
    const float* W0  = (const float*)d_in[3];
    const float* b0  = (const float*)d_in[4];
    const float* W1  = (const float*)d_in[5];
    const float* b1  = (const float*)d_in[6];
    const float* W2  = (const float*)d_in[7];
    const float* b2  = (const float*)d_in[8];
    const float* Wf1 = (const float*)d_in[9];
    const float* bf1 = (const float*)d_in[10];
    const float* Wf2 = (const float*)d_in[11];
    const float* bf2 = (const float*)d_in[12];

    const int N = in_sizes[0] / 3;
    const int E = in_sizes[1] / 2;
    const int G = out_size;           // 512 graphs

    // workspace partition (256B aligned): ~110 MB total
    char* p = (char*)d_ws;
    auto alloc = [&](size_t bytes) -> void* {
        void* r = (void*)p;
        p += (bytes + 255) & ~(size_t)255;
        return r;
    };
    int*   degi   = (int*)  alloc((size_t)N * 4);
    float* dis    = (float*)alloc((size_t)N * 4);
    float* norm   = (float*)alloc((size_t)E * 4);
    float* bufA   = (float*)alloc((size_t)N * HID * 4);   // H / agg (ping-pong)
    float* bufB   = (float*)alloc((size_t)N * HID * 4);   // T = H @ W
    float* pooled = (float*)alloc((size_t)G * HID * 4);
    int*   cnt    = (int*)  alloc((size_t)G * 4);
    float* dense1 = (float*)alloc((size_t)G * DENSE_DIM * 4);

    const int TB = 256;
    const int gN   = (N + TB - 1) / TB;
    const int gE   = (E + TB - 1) / TB;
    const int gNF4 = (int)(((size_t)N * (HID / 4) + TB - 1) / TB);   // float4 grids
    const int gGem = (N + 127) / 128;
    const int gSc  = (E + 7) / 8;

    // degree -> D^{-1/2} -> per-edge norm
    hipMemsetAsync(degi, 0, (size_t)N * 4, stream);
    k_deg <<<gE, TB, 0, stream>>>(ei + E, degi, E);
    k_dis <<<gN, TB, 0, stream>>>(degi, dis, N);
    k_norm<<<gE, TB, 0, stream>>>(ei, dis, norm, E);

    // layer 0 (Fin=3, scalar transform)
    k_lin3<<<gNF4, TB, 0, stream>>>(x, W0, bufB, N);
    hipMemsetAsync(bufA, 0, (size_t)N * HID * 4, stream);
    k_scatter <<<gSc, TB, 0, stream>>>(ei, ei + E, norm, bufB, bufA, E);
    k_finalize<<<gNF4, TB, 0, stream>>>(bufA, bufB, dis, b0, N);

    // layer 1 (WMMA fp32 GEMM)
    k_gemm128_wmma<<<gGem, TB, 0, stream>>>(bufA, W1, bufB, N);
    hipMemsetAsync(bufA, 0, (size_t)N * HID * 4, stream);
    k_scatter <<<gSc, TB, 0, stream>>>(ei, ei + E, norm, bufB, bufA, E);
    k_finalize<<<gNF4, TB, 0, stream>>>(bufA, bufB, dis, b1, N);

    // layer 2
    k_gemm128_wmma<<<gGem, TB, 0, stream>>>(bufA, W2, bufB, N);
    hipMemsetAsync(bufA, 0, (size_t)N * HID * 4, stream);
    k_scatter <<<gSc, TB, 0, stream>>>(ei, ei + E, norm, bufB, bufA, E);
    k_finalize<<<gNF4, TB, 0, stream>>>(bufA, bufB, dis, b2, N);

    // global mean pool + MLP head
    hipMemsetAsync(pooled, 0, (size_t)G * HID * 4, stream);
    hipMemsetAsync(cnt, 0, (size_t)G * 4, stream);
    k_pool_cnt<<<gN, TB, 0, stream>>>(bat, cnt, N);
    k_pool_sum<<<gNF4, TB, 0, stream>>>(bat, bufA, pooled, N);
    k_dense1<<<(G * DENSE_DIM + TB - 1) / TB, TB, 0, stream>>>(pooled, cnt, Wf1, bf1, dense1, G);
    k_dense2<<<(G + TB - 1) / TB, TB, 0, stream>>>(dense1, Wf2, bf2, (float*)d_out, G);
}